// opt_kde_4252017623491
// MI455X (gfx1250) — compile-verified
//
#include <hip/hip_runtime.h>
#include <hip/hip_bf16.h>

// Problem constants (from reference): E=4, L=8, S=512, F=32, SAMPLE=512
#define EE 4
#define LL 8
#define SS 512
#define FF 32
#define NCOL (EE*LL*FF)          // 1024 (e,l,f) columns
#define NROW (EE*LL*SS)          // 16384 gemm rows
#define NPTS 512                  // grid points

typedef __attribute__((ext_vector_type(2))) float v2f;
typedef __attribute__((ext_vector_type(8))) float v8f;

// ---------------- K1: per-(e,l,f) std over S (ddof=1) -----------------------
__global__ void __launch_bounds__(32) k1_colstd(const float* __restrict__ matrix,
                                                float* __restrict__ stdcol) {
  int col = blockIdx.x;                 // el*32 + f
  int el  = col >> 5, f = col & 31;
  const float* base = matrix + (size_t)el * SS * FF + f;
  int lane = threadIdx.x;
  float s = 0.f, s2 = 0.f;
  for (int i = lane; i < SS; i += 32) {
    float v = base[(size_t)i * FF];
    s += v; s2 += v * v;
  }
  for (int m = 16; m >= 1; m >>= 1) {
    s  += __shfl_xor(s,  m, 32);
    s2 += __shfl_xor(s2, m, 32);
  }
  if (lane == 0) {
    float mean = s * (1.f / (float)SS);
    float var  = (s2 - (float)SS * mean * mean) * (1.f / (float)(SS - 1));
    stdcol[col] = sqrtf(fmaxf(var, 0.f));
  }
}

// ---------------- K2: scalar std, bandwidth, constants ----------------------
__global__ void __launch_bounds__(32) k2_scalars(const float* __restrict__ stdcol,
                                                 float* __restrict__ sc) {
  int lane = threadIdx.x;
  float s = 0.f;
  for (int i = lane; i < NCOL; i += 32) s += stdcol[i];
  for (int m = 16; m >= 1; m >>= 1) s += __shfl_xor(s, m, 32);
  if (lane == 0) {
    float stdv  = s * (1.f / (float)NCOL);
    float ratio = stdv / stdv;                     // mirrors std(m/std).mean()
    float bw    = 1.06f * 0.28717458874925886f * ratio;  // 1.06 * S^(-1/5)
    float inv2bw2 = 0.5f / (bw * bw);
    sc[0] = stdv;
    sc[1] = bw;
    sc[2] = inv2bw2;
    sc[3] = 2.5066282746310002f * bw;              // sqrt(2*pi)*bw (divisor)
    sc[8] = -inv2bw2 * 1.4426950408889634f;        // fold log2(e) for v_exp_f32
  }
}

// ---------------- K3: proj GEMM via V_WMMA_F32_16X16X4_F32 ------------------
// One wave per 16-row tile of (e*l*s); two N=16 accumulators share the A frag.
// proj stored transposed as (el, g, s) so KDE column loads are contiguous.
__global__ void __launch_bounds__(32) k3_proj_wmma(const float* __restrict__ matrix,
                                                   const float* __restrict__ params,
                                                   const float* __restrict__ sc,
                                                   float* __restrict__ projT,
                                                   float* __restrict__ bmin,
                                                   float* __restrict__ bmax) {
  const float invstd = 1.0f / sc[0];
  int b     = blockIdx.x;          // 1024 tiles of 16 rows
  int lane  = threadIdx.x;         // wave32, EXEC all ones (WMMA requirement)
  int rbase = b * 16;
  int hi = lane >> 4;              // K-half selector per ISA A/B layout
  int lo = lane & 15;
  int row = rbase + lo;            // A: lanes 0-15 M=lo (K0,K1); 16-31 M=lo (K2,K3)
  v8f c0 = {}; v8f c1 = {};
#pragma unroll
  for (int k = 0; k < 8; ++k) {    // K=32 = 8 x K4 WMMA steps
    int koff = 4 * k + 2 * hi;
    v2f a;  a.x  = matrix[(size_t)row * FF + koff]     * invstd;
            a.y  = matrix[(size_t)row * FF + koff + 1] * invstd;
    v2f b0; b0.x = params[koff * FF + lo];       b0.y = params[(koff + 1) * FF + lo];
    v2f b1; b1.x = params[koff * FF + 16 + lo];  b1.y = params[(koff + 1) * FF + 16 + lo];
    c0 = __builtin_amdgcn_wmma_f32_16x16x4_f32(false, a, false, b0, (short)0, c0, false, false);
    c1 = __builtin_amdgcn_wmma_f32_16x16x4_f32(false, a, false, b1, (short)0, c1, false, false);
  }
  int el = rbase >> 9;             // rbase / 512 (tiles never straddle an (e,l))
  float mn = 3.402823466e38f, mx = -3.402823466e38f;
#pragma unroll
  for (int j = 0; j < 8; ++j) {    // C/D layout: VGPR j, lanes<16 M=j, >=16 M=8+j
    int r = rbase + j + 8 * hi;
    int s = r & (SS - 1);
    projT[((size_t)(el * FF + lo))      * SS + s] = c0[j];
    projT[((size_t)(el * FF + 16 + lo)) * SS + s] = c1[j];
    mn = fminf(mn, fminf(c0[j], c1[j]));
    mx = fmaxf(mx, fmaxf(c0[j], c1[j]));
  }
  for (int m = 16; m >= 1; m >>= 1) {
    mn = fminf(mn, __shfl_xor(mn, m, 32));
    mx = fmaxf(mx, __shfl_xor(mx, m, 32));
  }
  if (lane == 0) { bmin[b] = mn; bmax[b] = mx; }
}

// ---------------- K4: global min/max -> grid parameters ---------------------
__global__ void __launch_bounds__(32) k4_minmax(const float* __restrict__ bmin,
                                                const float* __restrict__ bmax,
                                                float* __restrict__ sc) {
  int lane = threadIdx.x;
  float mn = 3.402823466e38f, mx = -3.402823466e38f;
  for (int i = lane; i < 1024; i += 32) {
    mn = fminf(mn, bmin[i]); mx = fmaxf(mx, bmax[i]);
  }
  for (int m = 16; m >= 1; m >>= 1) {
    mn = fminf(mn, __shfl_xor(mn, m, 32));
    mx = fmaxf(mx, __shfl_xor(mx, m, 32));
  }
  if (lane == 0) {
    sc[4] = mn;                          // left
    sc[5] = mx;                          // right
    sc[6] = (mx - mn) / (float)NPTS;     // delta (reference divides by SAMPLE)
    sc[7] = (mx - mn) / (float)(NPTS-1); // linspace step
  }
}

// ---------------- K5: KDE main loop (268M v_exp_f32) ------------------------
// One block per (e,l,f) column; column staged in LDS via async-to-LDS copy.
__global__ void __launch_bounds__(256) k5_kde(const float* __restrict__ projT,
                                              const int* __restrict__ data_len,
                                              const float* __restrict__ sc,
                                              float* __restrict__ red) {
  __shared__ float smcol[SS];
  int col = blockIdx.x;                  // el*32 + f
  int el  = col >> 5;
  const float* colp = projT + (size_t)col * SS;
  int tid = threadIdx.x;

  // CDNA5 async global->LDS copy: 128 lanes x b128 = 2KB column, ASYNCcnt path
  if (tid < 128) {
    unsigned ldsa = (unsigned)(size_t)(&smcol[tid * 4]);   // low 32b = LDS offset
    const float* gp = colp + tid * 4;
    asm volatile("global_load_async_to_lds_b128 %0, %1, off"
                 :: "v"(ldsa), "v"(gp) : "memory");
  }
  asm volatile("s_wait_asynccnt 0" ::: "memory");
  __syncthreads();

  float left  = sc[4];
  float gstep = sc[7];
  float k2    = sc[8];                    // -log2(e)/(2*bw^2)
  float lenf  = (float)data_len[el];
  float pad   = (float)SS - lenf;
  float invlen = 1.f / lenf;

#pragma unroll
  for (int tt = 0; tt < 2; ++tt) {
    int t = tid + tt * 256;
    float g = left + (float)t * gstep;
    float acc = 0.f;
#pragma unroll 4
    for (int s = 0; s < SS; ++s) {
      float d = smcol[s] - g;
      acc += __builtin_amdgcn_exp2f(k2 * d * d);   // one v_exp_f32 per term
    }
    float corr = pad * __builtin_amdgcn_exp2f(k2 * g * g);
    red[(size_t)col * NPTS + t] = (acc - corr) * invlen;
  }
}

// ---------------- K6: pairwise L1 over env pairs, per (l,f) -----------------
__global__ void __launch_bounds__(128) k6_pairs(const float* __restrict__ red,
                                                const float* __restrict__ sc,
                                                float* __restrict__ out) {
  __shared__ float rbuf[128];
  int lf = blockIdx.x;                   // l*32 + f  (256 blocks)
  int l = lf >> 5, f = lf & 31;
  int tid = threadIdx.x;
  const int  pe1[6]     = {0, 0, 0, 1, 1, 2};
  const int  pe2[6]     = {1, 2, 3, 2, 3, 3};
  const bool istrain[6] = {true, true, false, true, false, false};
  float trainmax = 0.f, testmax = 0.f;   // distances are non-negative
  for (int p = 0; p < 6; ++p) {
    const float* ra = red + (size_t)((pe1[p] * LL + l) * FF + f) * NPTS;
    const float* rb = red + (size_t)((pe2[p] * LL + l) * FF + f) * NPTS;
    float s = 0.f;
    for (int t = tid; t < NPTS; t += 128) s += fabsf(ra[t] - rb[t]);
    rbuf[tid] = s;
    __syncthreads();
    for (int off = 64; off >= 1; off >>= 1) {
      if (tid < off) rbuf[tid] += rbuf[tid + off];
      __syncthreads();
    }
    float tot = rbuf[0];
    __syncthreads();
    testmax = fmaxf(testmax, tot);
    if (istrain[p]) trainmax = fmaxf(trainmax, tot);
  }
  if (tid == 0) {
    float scale = sc[6] / (2.f * sc[3]);  // delta / (2 * divisor)
    out[lf]       = trainmax * scale;     // train_results (L,F)
    out[256 + lf] = testmax  * scale;     // test_results  (L,F)
  }
}

// ---------------- K7: max over L, mean over F -> scalars --------------------
__global__ void __launch_bounds__(32) k7_final(float* __restrict__ out) {
  int f = threadIdx.x;                   // 32 lanes = F
  float trmax = 0.f, temax = 0.f;
  for (int l = 0; l < LL; ++l) {
    trmax = fmaxf(trmax, out[l * FF + f]);
    temax = fmaxf(temax, out[256 + l * FF + f]);
  }
  float trs = trmax, tes = temax;
  for (int m = 16; m >= 1; m >>= 1) {
    trs += __shfl_xor(trs, m, 32);
    tes += __shfl_xor(tes, m, 32);
  }
  if (f == 0) {
    out[512] = trs * (1.f / (float)FF);  // train_dis
    out[513] = tes * (1.f / (float)FF);  // test_dis
  }
}

extern "C" void kernel_launch(void* const* d_in, const int* in_sizes, int n_in,
                              void* d_out, int out_size, void* d_ws, size_t ws_size,
                              hipStream_t stream) {
  const float* matrix   = (const float*)d_in[0];   // (E,L,S,F)
  const float* params   = (const float*)d_in[1];   // (F,F)
  const int*   data_len = (const int*)d_in[2];     // (E,L)
  float* out = (float*)d_out;                      // 256 + 256 + 1 + 1
  float* ws  = (float*)d_ws;

  float* sc     = ws;                  // 16 scalars
  float* stdcol = ws + 16;             // 1024
  float* bmin   = ws + 16 + 1024;      // 1024
  float* bmax   = ws + 16 + 2048;      // 1024
  float* projT  = ws + 16 + 3072;      // 524288  (e,l,g,s)
  float* red    = projT + (size_t)NCOL * SS;  // 524288  (e,l,f,t)

  k1_colstd   <<<NCOL, 32, 0, stream>>>(matrix, stdcol);
  k2_scalars  <<<1,    32, 0, stream>>>(stdcol, sc);
  k3_proj_wmma<<<NROW / 16, 32, 0, stream>>>(matrix, params, sc, projT, bmin, bmax);
  k4_minmax   <<<1,    32, 0, stream>>>(bmin, bmax, sc);
  k5_kde      <<<NCOL, 256, 0, stream>>>(projT, data_len, sc, red);
  k6_pairs    <<<256, 128, 0, stream>>>(red, sc, out);
  k7_final    <<<1,    32, 0, stream>>>(out);
}